// TAGCN_54881092108447
// MI455X (gfx1250) — compile-verified
//
#include <hip/hip_runtime.h>
#include <math.h>

// ---- problem constants (match reference) ----
#define NNODES 50000
#define NEDGES 800000
#define FIN    128
#define HID    128
#define FOUT   64
#define KDIM   512   // (K+1) * feature = 4 * 128, GEMM reduction dim

typedef float v2f __attribute__((ext_vector_type(2)));
typedef float v4f __attribute__((ext_vector_type(4)));
typedef float v8f __attribute__((ext_vector_type(8)));

__device__ __forceinline__ void atomic_add_f32(float* p, float v) {
  __hip_atomic_fetch_add(p, v, __ATOMIC_RELAXED, __HIP_MEMORY_SCOPE_AGENT);
}

// ---------------- utility kernels ----------------
__global__ void tag_zero4(v4f* __restrict__ p, long n4) {
  long i = (long)blockIdx.x * blockDim.x + threadIdx.x;
  if (i < n4) p[i] = (v4f){0.f, 0.f, 0.f, 0.f};
}

__global__ void tag_zero(float* __restrict__ p, long n) {
  long i = (long)blockIdx.x * blockDim.x + threadIdx.x;
  if (i < n) p[i] = 0.0f;
}

__global__ void tag_deg(const long long* __restrict__ ei, float* __restrict__ deg) {
  int e = blockIdx.x * blockDim.x + threadIdx.x;
  if (e < NEDGES) {
    int c = (int)ei[NEDGES + e];           // col = targets
    atomic_add_f32(deg + c, 1.0f);
  }
}

__global__ void tag_dinv(float* __restrict__ d) {
  int i = blockIdx.x * blockDim.x + threadIdx.x;
  if (i < NNODES) {
    float v = d[i];
    d[i] = (v > 0.0f) ? rsqrtf(fmaxf(v, 1.0f)) : 0.0f;
  }
}

__global__ void tag_norm(const long long* __restrict__ ei,
                         const float* __restrict__ dinv,
                         float* __restrict__ nrm) {
  int e = blockIdx.x * blockDim.x + threadIdx.x;
  if (e < NEDGES) {
    int r = (int)ei[e];
    int c = (int)ei[NEDGES + e];
    nrm[e] = dinv[r] * dinv[c];
  }
}

// copy [N,128] row-major source into columns [0,128) of the [N,512] concat buffer
__global__ void tag_copy_to_concat(const float* __restrict__ src, float* __restrict__ dst) {
  long i = (long)blockIdx.x * blockDim.x + threadIdx.x;   // one thread per 4 floats
  if (i < (long)NNODES * (FIN / 4)) {
    long n = i >> 5;
    int  f = (int)(i & 31) * 4;
    *(v4f*)(dst + n * KDIM + f) = *(const v4f*)(src + n * FIN + f);
  }
}

// scatter-add SpMM: out[col,:] += norm[e] * in[row,:]; in/out are column slabs of
// the concat buffer (row stride KDIM). One thread per (edge, 4 features):
// vectorized gather, per-dword f32 atomics on the scatter side.
__global__ void tag_spmm(const long long* __restrict__ ei,
                         const float* __restrict__ nrm,
                         const float* __restrict__ in,
                         float* __restrict__ out) {
  long t = (long)blockIdx.x * blockDim.x + threadIdx.x;
  if (t >= (long)NEDGES * (FIN / 4)) return;
  int e = (int)(t >> 5);              // 32 threads cover one edge's 128 features
  int f = (int)(t & 31) * 4;
  int r = (int)ei[e];
  int c = (int)ei[NEDGES + e];
  float nv = nrm[e];
  v4f v = *(const v4f*)(in + (long)r * KDIM + f);
  float* o = out + (long)c * KDIM + f;
  atomic_add_f32(o + 0, nv * v.x);
  atomic_add_f32(o + 1, nv * v.y);
  atomic_add_f32(o + 2, nv * v.z);
  atomic_add_f32(o + 3, nv * v.w);
}

// ---------------- WMMA fp32 GEMM ----------------
// out[M,CC] = A[M,KDIM] @ B[KDIM,CC] + bias (+ optional ReLU), full fp32 via
// V_WMMA_F32_16X16X4_F32. blockDim.x = (CC/16)*32; each wave owns a 16-wide
// column tile and NT*16 rows (NT accumulators reuse each B fragment NT times).
// NT is compile-time -> branch-free inner loop; single base pointers with
// immediate tile offsets keep all loads on the GLOBAL path.
template <int NT, int CC, bool RELU>
__global__ __launch_bounds__(256) void tag_gemm_wmma(
    const float* __restrict__ A, const float* __restrict__ B,
    const float* __restrict__ bias, float* __restrict__ out, long mstart) {

  const int lane  = threadIdx.x & 31;
  const int wave  = threadIdx.x >> 5;
  const int n0    = wave * 16;           // column tile base
  const int nlane = lane & 15;
  const int half  = lane >> 4;           // 0: K pair {0,1}, 1: K pair {2,3}
  const long mbase = mstart + (long)blockIdx.x * (NT * 16);

  v8f acc[NT];
#pragma unroll
  for (int t = 0; t < NT; ++t) acc[t] = (v8f){};

  // A layout (16x4 fp32): lanes 0-15 -> K {kb,kb+1}, lanes 16-31 -> K {kb+2,kb+3};
  // contiguous K pair -> single b64 load. One base pointer; tiles addressed via
  // compile-time immediate offsets (t * 16 * KDIM floats = 32768*t bytes < 2^23).
  const float* abase = A + (mbase + nlane) * (long)KDIM + 2 * half;

  // B layout (4x16 fp32): V0 = row kk, V1 = row kk+1, column n0+nlane.
  const float* bbase = B + (long)(2 * half) * CC + n0 + nlane;

#pragma unroll 4
  for (int kb = 0; kb < KDIM; kb += 4) {
    v2f b;
    b.x = bbase[0];
    b.y = bbase[CC];
    v2f a[NT];
#pragma unroll
    for (int t = 0; t < NT; ++t)
      a[t] = *(const v2f*)(abase + t * (16 * KDIM));
#pragma unroll
    for (int t = 0; t < NT; ++t)
      acc[t] = __builtin_amdgcn_wmma_f32_16x16x4_f32(false, a[t], false, b,
                                                     (short)0, acc[t], false, false);
    abase += 4;
    bbase += 4 * CC;
  }

  // C/D layout: VGPR r -> row (mt + r) for lanes 0-15, (mt + r + 8) for lanes 16-31
  const int   col = n0 + nlane;
  const float bv  = bias[col];
#pragma unroll
  for (int t = 0; t < NT; ++t) {
    long mt = mbase + t * 16;
#pragma unroll
    for (int r = 0; r < 8; ++r) {
      long rowi = mt + r + 8 * half;
      float v = acc[t][r] + bv;
      if (RELU) v = fmaxf(v, 0.0f);
      out[rowi * (long)CC + col] = v;
    }
  }
}

// ---------------- log-softmax over 64 cols, one wave32 per node ----------------
__global__ void tag_logsoftmax(float* __restrict__ out) {
  int gwave = (int)(((long)blockIdx.x * blockDim.x + threadIdx.x) >> 5);
  int lane  = threadIdx.x & 31;
  if (gwave >= NNODES) return;                // wave-uniform guard
  float* row = out + (long)gwave * FOUT;
  float v0 = row[lane];
  float v1 = row[lane + 32];
  float m = fmaxf(v0, v1);
  for (int off = 16; off > 0; off >>= 1) m = fmaxf(m, __shfl_xor(m, off, 32));
  float s = expf(v0 - m) + expf(v1 - m);
  for (int off = 16; off > 0; off >>= 1) s += __shfl_xor(s, off, 32);
  float l = m + logf(s);
  row[lane]      = v0 - l;
  row[lane + 32] = v1 - l;
}

// ---------------- launcher ----------------
extern "C" void kernel_launch(void* const* d_in, const int* in_sizes, int n_in,
                              void* d_out, int out_size, void* d_ws, size_t ws_size,
                              hipStream_t stream) {
  const float*     x  = (const float*)d_in[0];
  const long long* ei = (const long long*)d_in[1];  // int64 [2, E]
  const float*     W1 = (const float*)d_in[2];      // [4,128,128] == [512,128] row-major
  const float*     b1 = (const float*)d_in[3];
  const float*     W2 = (const float*)d_in[4];      // [4,128,64]  == [512,64]  row-major
  const float*     b2 = (const float*)d_in[5];
  float* outp = (float*)d_out;

  // workspace carve-up (all fp32): norm[E] | dinv[N] | h1[N*128] | concat[N*512]
  float* nrm    = (float*)d_ws;
  float* dinv   = nrm + NEDGES;
  float* h1     = dinv + NNODES;
  float* concat = h1 + (long)NNODES * HID;

  const int T = 256;
  const long concat_q   = (long)NNODES * KDIM / 4;      // float4 count
  const long nodefeat_q = (long)NNODES * (FIN / 4);
  const long edgefeat_q = (long)NEDGES * (FIN / 4);

  // ---- normalization coefficients ----
  tag_zero<<<(NNODES + T - 1) / T, T, 0, stream>>>(dinv, NNODES);
  tag_deg<<<(NEDGES + T - 1) / T, T, 0, stream>>>(ei, dinv);
  tag_dinv<<<(NNODES + T - 1) / T, T, 0, stream>>>(dinv);
  tag_norm<<<(NEDGES + T - 1) / T, T, 0, stream>>>(ei, dinv, nrm);

  const int spmm_grid = (int)((edgefeat_q + T - 1) / T);
  const int zero_grid = (int)((concat_q + T - 1) / T);
  const int copy_grid = (int)((nodefeat_q + T - 1) / T);

  // 3125 M-tiles of 16 rows = 781 full 64-row panels + 1 tail tile
  const int  full_panels = (NNODES / 16) / 4;           // 781
  const long tail_start  = (long)full_panels * 64;      // 49984

  // ---- layer 1: hops into concat, fused GEMM (relu) -> h1 ----
  tag_zero4<<<zero_grid, T, 0, stream>>>((v4f*)concat, concat_q);
  tag_copy_to_concat<<<copy_grid, T, 0, stream>>>(x, concat);
  tag_spmm<<<spmm_grid, T, 0, stream>>>(ei, nrm, concat + 0 * FIN, concat + 1 * FIN);
  tag_spmm<<<spmm_grid, T, 0, stream>>>(ei, nrm, concat + 1 * FIN, concat + 2 * FIN);
  tag_spmm<<<spmm_grid, T, 0, stream>>>(ei, nrm, concat + 2 * FIN, concat + 3 * FIN);
  tag_gemm_wmma<4, HID, true><<<full_panels, (HID / 16) * 32, 0, stream>>>(
      concat, W1, b1, h1, 0);
  tag_gemm_wmma<1, HID, true><<<1, (HID / 16) * 32, 0, stream>>>(
      concat, W1, b1, h1, tail_start);

  // ---- layer 2: hops on h1, fused GEMM -> d_out, then log-softmax ----
  tag_zero4<<<zero_grid, T, 0, stream>>>((v4f*)concat, concat_q);
  tag_copy_to_concat<<<copy_grid, T, 0, stream>>>(h1, concat);
  tag_spmm<<<spmm_grid, T, 0, stream>>>(ei, nrm, concat + 0 * FIN, concat + 1 * FIN);
  tag_spmm<<<spmm_grid, T, 0, stream>>>(ei, nrm, concat + 1 * FIN, concat + 2 * FIN);
  tag_spmm<<<spmm_grid, T, 0, stream>>>(ei, nrm, concat + 2 * FIN, concat + 3 * FIN);
  tag_gemm_wmma<4, FOUT, false><<<full_panels, (FOUT / 16) * 32, 0, stream>>>(
      concat, W2, b2, outp, 0);
  tag_gemm_wmma<1, FOUT, false><<<1, (FOUT / 16) * 32, 0, stream>>>(
      concat, W2, b2, outp, tail_start);

  tag_logsoftmax<<<(NNODES * 32 + T - 1) / T, T, 0, stream>>>(outp);
}